// QuantumRegressor_61624190763551
// MI455X (gfx1250) — compile-verified
//
#include <hip/hip_runtime.h>
#include <hip/hip_bf16.h>

// CDNA5 / gfx1250 WMMA types
typedef __attribute__((ext_vector_type(2))) float v2f;
typedef __attribute__((ext_vector_type(8))) float v8f;

#define NQ 12
#define NLAYERS 4
#define BATCH 1024
#define NSTATE 4096   // 2^12
#define TPB 256       // 8 wave32 per workgroup

// ---------------------------------------------------------------------------
// Kernel 1: build fused 4-qubit gate matrices (batch-shared weights).
// For each (layer l, group g) with wires 4g..4g+3, compute the 16x16 complex
// matrix  U = kron(Rot(w[l,4g]), Rot(w[l,4g+1]), Rot(w[l,4g+2]), Rot(w[l,4g+3]))
// Rot(phi,theta,omega) = RZ(omega) RY(theta) RZ(phi)  (PennyLane convention).
// Stored row-major: Ur/Ui[(l*3+g)*256 + i*16 + j].
// ---------------------------------------------------------------------------
__global__ __launch_bounds__(256) void build_gates_kernel(
    const float* __restrict__ weights,  // (4,12,3)
    float* __restrict__ Ur, float* __restrict__ Ui) {
  int tid = threadIdx.x;
  int i = tid >> 4;    // output row 0..15
  int j = tid & 15;    // output col 0..15
  for (int l = 0; l < NLAYERS; ++l) {
    for (int g = 0; g < 3; ++g) {
      float re = 1.0f, im = 0.0f;
      #pragma unroll
      for (int q = 0; q < 4; ++q) {
        int w = g * 4 + q;
        const float* wp = weights + (l * NQ + w) * 3;
        float phi = wp[0], th = wp[1], om = wp[2];
        float c = cosf(0.5f * th), s = sinf(0.5f * th);
        float ap = 0.5f * (phi + om);   // m00 = e^{-i ap} c ; m11 = e^{+i ap} c
        float am = 0.5f * (phi - om);   // m01 = -e^{+i am} s ; m10 = e^{-i am} s
        int bi = (i >> (3 - q)) & 1;
        int bj = (j >> (3 - q)) & 1;
        float er, ei;
        if (bi == 0 && bj == 0)      { er =  cosf(ap) * c; ei = -sinf(ap) * c; }
        else if (bi == 0 && bj == 1) { er = -cosf(am) * s; ei = -sinf(am) * s; }
        else if (bi == 1 && bj == 0) { er =  cosf(am) * s; ei = -sinf(am) * s; }
        else                         { er =  cosf(ap) * c; ei =  sinf(ap) * c; }
        float nr = re * er - im * ei;
        float ni = re * ei + im * er;
        re = nr; im = ni;
      }
      int base = (l * 3 + g) * 256 + tid;
      Ur[base] = re;
      Ui[base] = im;
    }
  }
}

// ---------------------------------------------------------------------------
// Kernel 2: one workgroup per batch element. State (4096 complex) lives in LDS.
//  - RY embedding: per-batch kron'd 16x16 REAL matrices, applied with WMMA
//    (state stays real, so only the real chain is needed).
//  - Each layer: 3 fused 16x16 complex matmuls via v_wmma_f32_16x16x4_f32,
//    then the 12-CNOT permutation (composed XOR index map).
//  - PauliZ expvals folded into a per-amplitude coefficient, block reduction.
// ---------------------------------------------------------------------------
__global__ __launch_bounds__(TPB) void qsim_kernel(
    const float* __restrict__ x,       // (1024,12)
    const float* __restrict__ Ur,      // (12,16,16)
    const float* __restrict__ Ui,      // (12,16,16)
    const float* __restrict__ head_w,  // (1,12)
    const float* __restrict__ head_b,  // (1,)
    float* __restrict__ out) {         // (1024,)
  __shared__ float sre[NSTATE];
  __shared__ float sim_[NSTATE];
  const int b = blockIdx.x;
  const int tid = threadIdx.x;
  const int lane = tid & 31;
  const int wv = tid >> 5;         // 0..7
  const int ahi = lane >> 4;       // 0/1 : which K-half this lane holds
  const int nlo = lane & 15;       // row (A) / column (B,D) within tile

  // strides for viewing the state index as (axis j, tile t, col n):
  //   g=0 (wires 0-3, bits 11-8):  idx = j*256 + t*16  + n*1
  //   g=1 (wires 4-7, bits  7-4):  idx = j*16  + t*256 + n*1
  //   g=2 (wires 8-11,bits  3-0):  idx = j*1   + t*256 + n*16
  const int JS[3] = {256, 16, 1};
  const int TS[3] = {16, 256, 256};
  const int NS[3] = {1, 1, 16};

  // --- init |0..0> ---
  for (int i = tid; i < NSTATE; i += TPB) {
    sre[i] = (i == 0) ? 1.0f : 0.0f;
    sim_[i] = 0.0f;
  }
  __syncthreads();

  // --- AngleEmbedding: RY(x[b,w]) on wire w. The 4 RYs of each wire-group
  //     kron into a per-batch 16x16 REAL matrix; A-fragment entries are built
  //     in-register and applied with the real WMMA chain only (im stays 0). ---
  #pragma unroll
  for (int g = 0; g < 3; ++g) {
    const int js = JS[g], ts = TS[g], ns = NS[g];
    float c4[4], s4[4];
    #pragma unroll
    for (int q = 0; q < 4; ++q) {
      float a = 0.5f * x[b * NQ + g * 4 + q];
      c4[q] = cosf(a);
      s4[q] = sinf(a);
    }
    // RY = [[c,-s],[s,c]]; entry (row=nlo, col=kk+r) = prod_q m_q(bit_i,bit_j)
    v2f ar[4];
    #pragma unroll
    for (int k0 = 0; k0 < 4; ++k0) {
      int kk = k0 * 4 + ahi * 2;
      #pragma unroll
      for (int r = 0; r < 2; ++r) {
        int jcol = kk + r;
        float v = 1.0f;
        #pragma unroll
        for (int q = 0; q < 4; ++q) {
          int bi = (nlo >> (3 - q)) & 1;
          int bj = (jcol >> (3 - q)) & 1;
          float m = bi ? (bj ? c4[q] : s4[q]) : (bj ? -s4[q] : c4[q]);
          v *= m;
        }
        if (r == 0) ar[k0].x = v; else ar[k0].y = v;
      }
    }
    #pragma unroll
    for (int tt = 0; tt < 2; ++tt) {
      int t = wv + tt * 8;
      v2f br[4];
      #pragma unroll
      for (int k0 = 0; k0 < 4; ++k0) {
        int kk = k0 * 4 + ahi * 2;
        int idx = kk * js + t * ts + nlo * ns;
        br[k0].x = sre[idx];
        br[k0].y = sre[idx + js];
      }
      v8f dre = {};
      #pragma unroll
      for (int k0 = 0; k0 < 4; ++k0)
        dre = __builtin_amdgcn_wmma_f32_16x16x4_f32(
            false, ar[k0], false, br[k0], (short)0, dre, false, false);
      #pragma unroll
      for (int r = 0; r < 8; ++r) {
        int m = r + ahi * 8;
        sre[m * js + t * ts + nlo * ns] = dre[r];
      }
    }
    __syncthreads();
  }

  // --- StronglyEntanglingLayers ---
  for (int l = 0; l < NLAYERS; ++l) {
    #pragma unroll
    for (int g = 0; g < 3; ++g) {
      const int js = JS[g], ts = TS[g], ns = NS[g];
      const float* ur = Ur + (l * 3 + g) * 256;
      const float* ui = Ui + (l * 3 + g) * 256;

      // A fragments (gate matrix), shared by both tiles of this wave.
      // K-step k0 covers K in [4k0, 4k0+4); lane<16 holds K=4k0+{0,1},
      // lane>=16 holds K=4k0+{2,3}.  Row = lane%16.
      v2f ar[4], ai[4], nai[4];
      #pragma unroll
      for (int k0 = 0; k0 < 4; ++k0) {
        int kk = k0 * 4 + ahi * 2;
        ar[k0].x = ur[nlo * 16 + kk];
        ar[k0].y = ur[nlo * 16 + kk + 1];
        ai[k0].x = ui[nlo * 16 + kk];
        ai[k0].y = ui[nlo * 16 + kk + 1];
        nai[k0].x = -ai[k0].x;
        nai[k0].y = -ai[k0].y;
      }

      // Each wave transforms 2 independent 16x16 tiles (read set == write set,
      // so no barrier needed until the whole group finishes).
      #pragma unroll
      for (int tt = 0; tt < 2; ++tt) {
        int t = wv + tt * 8;
        v2f br[4], bi[4];
        #pragma unroll
        for (int k0 = 0; k0 < 4; ++k0) {
          int kk = k0 * 4 + ahi * 2;
          int idx = kk * js + t * ts + nlo * ns;
          br[k0].x = sre[idx];
          br[k0].y = sre[idx + js];
          bi[k0].x = sim_[idx];
          bi[k0].y = sim_[idx + js];
        }
        v8f dre = {};
        v8f dim = {};
        #pragma unroll
        for (int k0 = 0; k0 < 4; ++k0)
          dre = __builtin_amdgcn_wmma_f32_16x16x4_f32(
              false, ar[k0], false, br[k0], (short)0, dre, false, false);
        #pragma unroll
        for (int k0 = 0; k0 < 4; ++k0)
          dre = __builtin_amdgcn_wmma_f32_16x16x4_f32(
              false, nai[k0], false, bi[k0], (short)0, dre, false, false);
        #pragma unroll
        for (int k0 = 0; k0 < 4; ++k0)
          dim = __builtin_amdgcn_wmma_f32_16x16x4_f32(
              false, ar[k0], false, bi[k0], (short)0, dim, false, false);
        #pragma unroll
        for (int k0 = 0; k0 < 4; ++k0)
          dim = __builtin_amdgcn_wmma_f32_16x16x4_f32(
              false, ai[k0], false, br[k0], (short)0, dim, false, false);
        // D layout: VGPR r -> M = r + 8*ahi, N = lane%16
        #pragma unroll
        for (int r = 0; r < 8; ++r) {
          int m = r + ahi * 8;
          int idx = m * js + t * ts + nlo * ns;
          sre[idx] = dre[r];
          sim_[idx] = dim[r];
        }
      }
      __syncthreads();
    }

    // --- entangling CNOT ring: CNOT(w, (w+r)%12) for w=0..11, applied in
    //     order; r = (l % 11) + 1 == l+1 here (l < 4).
    //     Composed permutation: out[i] = in[c0(c1(...c11(i)...))]. ---
    int rr = l + 1;
    float tre[16], tim[16];
    #pragma unroll
    for (int k = 0; k < 16; ++k) {
      int i = tid + k * TPB;
      int src = i;
      #pragma unroll
      for (int w = NQ - 1; w >= 0; --w) {
        int cb = 11 - w;
        int tw = w + rr;
        if (tw >= NQ) tw -= NQ;          // (w+rr) % 12 without division
        int tb = 11 - tw;
        src ^= ((src >> cb) & 1) << tb;
      }
      tre[k] = sre[src];
      tim[k] = sim_[src];
    }
    __syncthreads();
    #pragma unroll
    for (int k = 0; k < 16; ++k) {
      int i = tid + k * TPB;
      sre[i] = tre[k];
      sim_[i] = tim[k];
    }
    __syncthreads();
  }

  // --- y[b] = sum_i |s_i|^2 * (sum_w head_w[w]*(1-2*bit_w(i))) + head_b ---
  // i = k*256 + tid  =>  coef(i) = chigh(k) + clow(tid)
  float hw[NQ];
  #pragma unroll
  for (int w = 0; w < NQ; ++w) hw[w] = head_w[w];
  float clow = 0.0f;
  #pragma unroll
  for (int w = 4; w < NQ; ++w)
    clow += ((tid >> (11 - w)) & 1) ? -hw[w] : hw[w];
  float partial = 0.0f;
  #pragma unroll
  for (int k = 0; k < 16; ++k) {
    int i = tid + k * TPB;
    float chigh = 0.0f;
    #pragma unroll
    for (int w = 0; w < 4; ++w)
      chigh += ((k >> (3 - w)) & 1) ? -hw[w] : hw[w];
    float pr = sre[i], pi = sim_[i];
    partial += (pr * pr + pi * pi) * (chigh + clow);
  }
  __syncthreads();
  sre[tid] = partial;  // reuse LDS for reduction
  __syncthreads();
  for (int s2 = TPB / 2; s2 > 0; s2 >>= 1) {
    if (tid < s2) sre[tid] += sre[tid + s2];
    __syncthreads();
  }
  if (tid == 0) out[b] = sre[0] + head_b[0];
}

// ---------------------------------------------------------------------------
extern "C" void kernel_launch(void* const* d_in, const int* in_sizes, int n_in,
                              void* d_out, int out_size, void* d_ws, size_t ws_size,
                              hipStream_t stream) {
  const float* x       = (const float*)d_in[0];  // (1024,12)
  const float* weights = (const float*)d_in[1];  // (4,12,3)
  const float* head_w  = (const float*)d_in[2];  // (1,12)
  const float* head_b  = (const float*)d_in[3];  // (1,)
  float* out = (float*)d_out;                    // (1024,)

  float* Ur = (float*)d_ws;                      // 12 * 256 floats
  float* Ui = Ur + 12 * 256;                     // 12 * 256 floats (24KB total)

  build_gates_kernel<<<1, 256, 0, stream>>>(weights, Ur, Ui);
  qsim_kernel<<<BATCH, TPB, 0, stream>>>(x, Ur, Ui, head_w, head_b, out);
}